// QKVAttention_89996744721172
// MI455X (gfx1250) — compile-verified
//
#include <hip/hip_runtime.h>
#include <math.h>
#include <stdint.h>

typedef __attribute__((ext_vector_type(2))) float v2f;
typedef __attribute__((ext_vector_type(8))) float v8f;

namespace {

constexpr int H = 8;
constexpr int L = 256;
constexpr int C = 32;
constexpr int THREADS = 512;   // 16 waves (wave32); each wave owns 16 query rows

__global__ __launch_bounds__(THREADS)
void tri_attn_kernel(const float* __restrict__ q,
                     const float* __restrict__ k,
                     const float* __restrict__ v,
                     const float* __restrict__ bias,
                     const float* __restrict__ gate,
                     float* __restrict__ out)
{
    __shared__ float v_lds[L * C];        // 32 KB: V for this (h,s1) pair
    __shared__ float p_lds[16][16][18];   // per-wave P re-layout tile, padded stride

    const int pair  = blockIdx.x;         // pair = h*L + s1
    const int h     = pair >> 8;
    const int s1    = pair & (L - 1);
    const int tid   = threadIdx.x;
    const int wid   = tid >> 5;           // 0..15
    const int lane  = tid & 31;
    const int lhalf = lane >> 4;          // 0 | 1
    const int l16   = lane & 15;

    const size_t poff = (size_t)pair * (L * C);
    const float* qp = q + poff;
    const float* kp = k + poff;
    const float* vp = v + poff;

    // ---- stage V into LDS via CDNA5 async copy (no VGPR round-trip) ----
    // 32 KB total; each of 512 threads moves 4 x 16B. INST_OFFSET is applied to
    // both the LDS and the global address, so one (lds, voffset) pair serves all 4.
    {
        const uint32_t lds_off = (uint32_t)(uintptr_t)v_lds + (uint32_t)tid * 16u;
        const uint32_t voff    = (uint32_t)tid * 16u;
        asm volatile("global_load_async_to_lds_b128 %0, %1, %2"
                     :: "v"(lds_off), "v"(voff), "s"(vp) : "memory");
        asm volatile("global_load_async_to_lds_b128 %0, %1, %2 offset:8192"
                     :: "v"(lds_off), "v"(voff), "s"(vp) : "memory");
        asm volatile("global_load_async_to_lds_b128 %0, %1, %2 offset:16384"
                     :: "v"(lds_off), "v"(voff), "s"(vp) : "memory");
        asm volatile("global_load_async_to_lds_b128 %0, %1, %2 offset:24576"
                     :: "v"(lds_off), "v"(voff), "s"(vp) : "memory");
        asm volatile("s_wait_asynccnt 0x0" ::: "memory");
    }
    __syncthreads();

    const int qbase = wid * 16;           // this wave's query rows

    // Per-lane base pointers; all further addressing is via constant immediates.
    const float* qb = qp + (size_t)(qbase + l16) * C + 2 * lhalf;
    const float* kb = kp + (size_t)l16 * C + 2 * lhalf;
    const float* bp = bias + ((size_t)(qbase + 8 * lhalf) * L + (size_t)l16) * H + h;

    // ---- load Q in WMMA f32 16x16x4 A-layout: lane = row l16, K-pair 2*lhalf ----
    v2f a[8];
    #pragma unroll
    for (int ks = 0; ks < 8; ++ks)
        a[ks] = *(const v2f*)(qb + ks * 4);

    // ---- S = Q*K^T + bias ----
    // Both the K B-tiles AND the bias accumulator-init are software-pipelined one
    // key-tile ahead: all loads for tile nt+1 issue before the WMMAs of tile nt,
    // so the wait before each WMMA group leaves a full tile of loads in flight.
    v8f s[16];
    v2f bb[2][8];
    v8f accN;
    #pragma unroll
    for (int ks = 0; ks < 8; ++ks)
        bb[0][ks] = *(const v2f*)(kb + ks * 4);
    #pragma unroll
    for (int r = 0; r < 8; ++r)
        accN[r] = bp[r * (L * H)];

    #pragma unroll
    for (int nt = 0; nt < 16; ++nt) {
        const int cur = nt & 1;
        v8f acc = accN;
        if (nt < 15) {
            #pragma unroll
            for (int r = 0; r < 8; ++r)
                accN[r] = bp[r * (L * H) + (nt + 1) * (16 * H)];
            #pragma unroll
            for (int ks = 0; ks < 8; ++ks)
                bb[cur ^ 1][ks] = *(const v2f*)(kb + (nt + 1) * 16 * C + ks * 4);
        }
        #pragma unroll
        for (int ks = 0; ks < 8; ++ks) {
            acc = __builtin_amdgcn_wmma_f32_16x16x4_f32(
                      false, a[ks], false, bb[cur][ks], (short)0, acc, false, false);
        }
        s[nt] = acc;
    }

    // ---- softmax over keys: row = fixed component r within a 16-lane half ----
    v8f mx = s[0];
    #pragma unroll
    for (int nt = 1; nt < 16; ++nt) {
        #pragma unroll
        for (int r = 0; r < 8; ++r) mx[r] = fmaxf(mx[r], s[nt][r]);
    }
    #pragma unroll
    for (int off = 1; off < 16; off <<= 1) {
        #pragma unroll
        for (int r = 0; r < 8; ++r)
            mx[r] = fmaxf(mx[r], __shfl_xor(mx[r], off, 32));
    }

    v8f rs;
    #pragma unroll
    for (int r = 0; r < 8; ++r) rs[r] = 0.0f;
    #pragma unroll
    for (int nt = 0; nt < 16; ++nt) {
        #pragma unroll
        for (int r = 0; r < 8; ++r) {
            const float p = __expf(s[nt][r] - mx[r]);
            s[nt][r] = p;
            rs[r]   += p;
        }
    }
    #pragma unroll
    for (int off = 1; off < 16; off <<= 1) {
        #pragma unroll
        for (int r = 0; r < 8; ++r)
            rs[r] += __shfl_xor(rs[r], off, 32);
    }

    // ---- O = P*V : re-layout P (C-layout -> A-layout) through per-wave LDS ----
    // Single LDS tile, register-double-buffered operands, pipelined one tile
    // ahead. DS ops are in-order per wave: reads(nt) complete before the
    // stores(nt+1) that overwrite the tile, so this is hazard-free.
    v8f o0, o1;
    #pragma unroll
    for (int r = 0; r < 8; ++r) { o0[r] = 0.0f; o1[r] = 0.0f; }

    float (*pst)[18] = p_lds[wid];
    const float* vbp = v_lds + 2 * lhalf * C + l16;   // per-lane V base in LDS

    v2f pa[2][4], vb0[2][4], vb1[2][4];
    // prologue: stage tile 0 and read its operands
    #pragma unroll
    for (int r = 0; r < 8; ++r)
        pst[r + 8 * lhalf][l16] = s[0][r];
    #pragma unroll
    for (int kk = 0; kk < 4; ++kk) {
        pa[0][kk] = *(const v2f*)&pst[l16][kk * 4 + 2 * lhalf];
        const int koff = (kk * 4) * C;
        vb0[0][kk].x = vbp[koff];
        vb0[0][kk].y = vbp[koff + C];
        vb1[0][kk].x = vbp[koff + 16];
        vb1[0][kk].y = vbp[koff + C + 16];
    }

    #pragma unroll
    for (int nt = 0; nt < 16; ++nt) {
        const int cur = nt & 1;
        if (nt < 15) {
            #pragma unroll
            for (int r = 0; r < 8; ++r)
                pst[r + 8 * lhalf][l16] = s[nt + 1][r];
            #pragma unroll
            for (int kk = 0; kk < 4; ++kk) {
                pa[cur ^ 1][kk] = *(const v2f*)&pst[l16][kk * 4 + 2 * lhalf];
                const int koff = ((nt + 1) * 16 + kk * 4) * C;
                vb0[cur ^ 1][kk].x = vbp[koff];
                vb0[cur ^ 1][kk].y = vbp[koff + C];
                vb1[cur ^ 1][kk].x = vbp[koff + 16];
                vb1[cur ^ 1][kk].y = vbp[koff + C + 16];
            }
        }
        #pragma unroll
        for (int kk = 0; kk < 4; ++kk) {
            o0 = __builtin_amdgcn_wmma_f32_16x16x4_f32(
                     false, pa[cur][kk], false, vb0[cur][kk], (short)0, o0, false, false);
            o1 = __builtin_amdgcn_wmma_f32_16x16x4_f32(
                     false, pa[cur][kk], false, vb1[cur][kk], (short)0, o1, false, false);
        }
    }

    // ---- normalize, gate, store (const-offset addressing off per-lane bases) ----
    const float* gp = gate + (((size_t)(s1 * L + qbase + 8 * lhalf)) * C + (size_t)l16) * H + h;
    float* op = out + poff + (size_t)(qbase + 8 * lhalf) * C + l16;
    #pragma unroll
    for (int r = 0; r < 8; ++r) {
        const float inv = 1.0f / rs[r];
        const float g0 = gp[r * (C * H)];
        const float g1 = gp[r * (C * H) + 16 * H];
        op[r * C]      = o0[r] * inv * g0;
        op[r * C + 16] = o1[r] * inv * g1;
    }
}

} // namespace

extern "C" void kernel_launch(void* const* d_in, const int* in_sizes, int n_in,
                              void* d_out, int out_size, void* d_ws, size_t ws_size,
                              hipStream_t stream) {
    (void)in_sizes; (void)n_in; (void)out_size; (void)d_ws; (void)ws_size;
    const float* q    = (const float*)d_in[0];
    const float* k    = (const float*)d_in[1];
    const float* v    = (const float*)d_in[2];
    const float* bias = (const float*)d_in[3];
    const float* gate = (const float*)d_in[4];
    float* out = (float*)d_out;

    tri_attn_kernel<<<H * L, THREADS, 0, stream>>>(q, k, v, bias, gate, out);
}